// s_prod_KernelChannels_76270029242959
// MI455X (gfx1250) — compile-verified
//
#include <hip/hip_runtime.h>

// Reference: B=4, C=16, H=W=256, K=3, PAD=1, MID=4.
// out = concat(s_out, cs_out), each [B][C][9][H][W] f32.
// s_out[b,c,ki*3+kj,h,w] = s[b,c,h+ki-1,w+kj-1] * s[b,c,h,w]  (zero-padded),
// except idx==4 which passes through s[b,c,h,w]. Same for cs.

typedef float v4f __attribute__((ext_vector_type(4)));

#define HH 256
#define WW 256
#define NPLANES 64          // B*C
#define HW (HH * WW)        // 65536
#define OUT_PLANE (9 * HW)  // 589824 floats per (b,c) per tensor
#define ROWS 4              // rows per 256-thread block

// Branch-free plane processing: all loads are unconditional (addresses clamped
// to valid memory); validity is applied with mask multiplies / selects only.
__device__ __forceinline__ void process_plane(const float* __restrict__ in_plane,
                                              float* __restrict__ ob,
                                              const int off[3],
                                              const float rmask[3],
                                              int dl, int dr,
                                              float lmask, float rmask_w) {
    v4f v[3];
    float L[3], R[3];
#pragma unroll
    for (int r = 0; r < 3; ++r) {
        const float* rp = in_plane + off[r];
        v4f t = *(const v4f*)rp;               // global_load_b128, always in-bounds
        float lraw = rp[dl];                   // dl = -1 or 0 (selected, in-bounds)
        float rraw = rp[dr];                   // dr = +4 or 0 (selected, in-bounds)
        v[r] = t * rmask[r];                   // zero out-of-range rows
        L[r] = lraw * (lmask * rmask[r]);
        R[r] = rraw * (rmask_w * rmask[r]);
    }
    const v4f mid = v[1];                      // rmask[1] == 1 always
#pragma unroll
    for (int ki = 0; ki < 3; ++ki) {
#pragma unroll
        for (int kj = 0; kj < 3; ++kj) {
            const int idx = ki * 3 + kj;
            v4f sv;
            if (kj == 0) {        // shift right: left halo enters
                sv.x = L[ki]; sv.y = v[ki].x; sv.z = v[ki].y; sv.w = v[ki].z;
            } else if (kj == 1) { // aligned
                sv = v[ki];
            } else {              // shift left: right halo enters
                sv.x = v[ki].y; sv.y = v[ki].z; sv.z = v[ki].w; sv.w = R[ki];
            }
            const v4f o = (idx == 4) ? sv : sv * mid;
            // 302 MB output stream, never re-read, larger than the 192 MB L2:
            // non-temporal store keeps it out of the way of the input stream.
            __builtin_nontemporal_store(o, (v4f*)(ob + idx * HW));
        }
    }
}

__global__ __launch_bounds__(256) void
s_prod_KernelChannels_76270029242959_kernel(const float* __restrict__ s,
                                            const float* __restrict__ cs,
                                            float* __restrict__ out) {
    const int tid   = threadIdx.x;
    const int strip = blockIdx.x;            // [0, NPLANES * HH/ROWS)
    const int plane = strip >> 6;            // 64 strips per plane
    const int h     = ((strip & 63) << 2) + (tid >> 6);
    const int w0    = (tid & 63) << 2;       // 4 floats per thread

    // Shared (tensor-independent) address math, computed once.
    const float lmask   = (w0 > 0)      ? 1.0f : 0.0f;
    const float rmask_w = (w0 < WW - 4) ? 1.0f : 0.0f;
    const int   dl      = (w0 > 0)      ? -1 : 0;   // halo offsets stay in-bounds
    const int   dr      = (w0 < WW - 4) ?  4 : 0;

    int   off[3];
    float rmask[3];
#pragma unroll
    for (int r = 0; r < 3; ++r) {
        const int row = h + r - 1;
        const bool ok = (r == 1) ? true : ((unsigned)row < (unsigned)HH);
        const int  rc = ok ? row : h;        // clamp to center row (always valid)
        off[r]   = rc * WW + w0;
        rmask[r] = ok ? 1.0f : 0.0f;
    }

    const float* sp = s  + plane * HW;
    const float* cp = cs + plane * HW;
    float* so = out + plane * OUT_PLANE + h * WW + w0;
    float* co = so + NPLANES * OUT_PLANE;

    // gfx1250 prefetch of the second tensor's rows while we process the first.
    __builtin_prefetch(cp + off[0], 0, 1);   // -> global_prefetch_b8
    __builtin_prefetch(cp + off[2], 0, 1);

    process_plane(sp, so, off, rmask, dl, dr, lmask, rmask_w);
    process_plane(cp, co, off, rmask, dl, dr, lmask, rmask_w);
}

extern "C" void kernel_launch(void* const* d_in, const int* in_sizes, int n_in,
                              void* d_out, int out_size, void* d_ws, size_t ws_size,
                              hipStream_t stream) {
    (void)in_sizes; (void)n_in; (void)out_size; (void)d_ws; (void)ws_size;
    const float* s  = (const float*)d_in[0];
    const float* cs = (const float*)d_in[1];
    float* out = (float*)d_out;

    const int blocks = NPLANES * (HH / ROWS);   // 4096
    s_prod_KernelChannels_76270029242959_kernel<<<blocks, 256, 0, stream>>>(s, cs, out);
}